// ThreeLayerGCNWithEdgeFeatures_29094108463688
// MI455X (gfx1250) — compile-verified
//
#include <hip/hip_runtime.h>

typedef float v2f __attribute__((ext_vector_type(2)));
typedef float v8f __attribute__((ext_vector_type(8)));

// ---------------------------------------------------------------------------
// Degree / normalization kernels
// ---------------------------------------------------------------------------
__global__ __launch_bounds__(256) void k_deg_init(float* __restrict__ deg, int n) {
    int i = blockIdx.x * 256 + threadIdx.x;
    if (i < n) deg[i] = 1.0f;   // self-loop
}

__global__ __launch_bounds__(256) void k_deg_count(const int* __restrict__ dst,
                                                   float* __restrict__ deg, int E) {
    int e = blockIdx.x * 256 + threadIdx.x;
    if (e < E) unsafeAtomicAdd(&deg[dst[e]], 1.0f);
}

__global__ __launch_bounds__(256) void k_dinv(const float* __restrict__ deg,
                                              float* __restrict__ dinv, int n) {
    int i = blockIdx.x * 256 + threadIdx.x;
    if (i < n) dinv[i] = rsqrtf(deg[i]);
}

// ---------------------------------------------------------------------------
// fp32 WMMA GEMM: H[M,N] = X[M,K] * W[K,N]
// One wave per 16x(16*NT) output strip: NT independent accumulator chains,
// each A fragment reused NT times.
// A layout (16x4 f32): lanes 0-15 -> K=0,1 ; lanes 16-31 -> K=2,3
// B layout (4x16 f32): VGPR0: K=koff, VGPR1: K=koff+1, col = lane&15
// C/D layout (16x16 f32): VGPR r -> row r (lanes 0-15) / row r+8 (lanes 16-31)
// ---------------------------------------------------------------------------
template<int K, int N, int NT>
__global__ __launch_bounds__(32) void k_gemm_wmma(const float* __restrict__ X,
                                                  const float* __restrict__ W,
                                                  float* __restrict__ H) {
    const int lane = threadIdx.x & 31;
    const int tm   = blockIdx.x * 16;
    const int tn   = blockIdx.y * (16 * NT);
    const int mrow = tm + (lane & 15);
    const int koff = (lane >> 4) << 1;       // 0 for lanes 0-15, 2 for lanes 16-31
    const int col  = tn + (lane & 15);

    v8f c[NT];
#pragma unroll
    for (int j = 0; j < NT; ++j) c[j] = (v8f){};

#pragma unroll 4
    for (int k0 = 0; k0 < K; k0 += 4) {
        v2f a = *(const v2f*)(X + (size_t)mrow * K + k0 + koff);  // 8B aligned pair
        const float* w0 = W + (size_t)(k0 + koff) * N + col;
#pragma unroll
        for (int j = 0; j < NT; ++j) {
            v2f b;
            b.x = w0[j * 16];
            b.y = w0[j * 16 + N];
            c[j] = __builtin_amdgcn_wmma_f32_16x16x4_f32(false, a, false, b,
                                                         (short)0, c[j], false, false);
        }
    }

    const int rbase = tm + ((lane >> 4) << 3);
#pragma unroll
    for (int j = 0; j < NT; ++j)
#pragma unroll
        for (int r = 0; r < 8; ++r)
            H[(size_t)(rbase + r) * N + col + j * 16] = c[j][r];
}

// ---------------------------------------------------------------------------
// agg[i,:] = h[i,:] * dinv[i]^2 + bias   (self-loop term + folded bias;
// also initializes the agg buffer, so no zero-fill pass is needed)
// ---------------------------------------------------------------------------
template<int F>
__global__ __launch_bounds__(256) void k_self_init(const float* __restrict__ h,
                                                   const float* __restrict__ dinv,
                                                   const float* __restrict__ bias,
                                                   float* __restrict__ agg, int n) {
    constexpr int F4 = F / 4;
    int i4 = blockIdx.x * 256 + threadIdx.x;
    if (i4 >= n * F4) return;
    int node = i4 / F4;
    int c4   = i4 - node * F4;
    float s = dinv[node];
    s *= s;
    float4 v = ((const float4*)h)[i4];
    float4 b = ((const float4*)bias)[c4];
    ((float4*)agg)[i4] = make_float4(fmaf(v.x, s, b.x), fmaf(v.y, s, b.y),
                                     fmaf(v.z, s, b.z), fmaf(v.w, s, b.w));
}

// ---------------------------------------------------------------------------
// Edge scatter-add: agg[dst] += h[src] * dinv[src]*dinv[dst]
// F/4 lanes cooperate on one edge; float4 gather + 4x global_atomic_add_f32
// ---------------------------------------------------------------------------
template<int F>
__global__ __launch_bounds__(256) void k_edge_agg(const float* __restrict__ h,
                                                  const int* __restrict__ src,
                                                  const int* __restrict__ dst,
                                                  const float* __restrict__ dinv,
                                                  float* __restrict__ agg, int E) {
    constexpr int LPE = F / 4;            // lanes per edge
    int t    = blockIdx.x * 256 + threadIdx.x;
    int e    = t / LPE;
    int lane = t - e * LPE;
    if (e >= E) return;
    int s = src[e], d = dst[e];
    float w = dinv[s] * dinv[d];
    float4 v = ((const float4*)h)[(size_t)s * LPE + lane];
    float* ap = agg + (size_t)d * F + lane * 4;
    unsafeAtomicAdd(ap + 0, v.x * w);
    unsafeAtomicAdd(ap + 1, v.y * w);
    unsafeAtomicAdd(ap + 2, v.z * w);
    unsafeAtomicAdd(ap + 3, v.w * w);
}

// ---------------------------------------------------------------------------
// In-place vectorized ReLU
// ---------------------------------------------------------------------------
__global__ __launch_bounds__(256) void k_relu(float* __restrict__ y, int total4) {
    int i = blockIdx.x * 256 + threadIdx.x;
    if (i >= total4) return;
    float4 v = ((float4*)y)[i];
    ((float4*)y)[i] = make_float4(fmaxf(v.x, 0.0f), fmaxf(v.y, 0.0f),
                                  fmaxf(v.z, 0.0f), fmaxf(v.w, 0.0f));
}

// ---------------------------------------------------------------------------
extern "C" void kernel_launch(void* const* d_in, const int* in_sizes, int n_in,
                              void* d_out, int out_size, void* d_ws, size_t ws_size,
                              hipStream_t stream) {
    const float* x  = (const float*)d_in[0];
    const int*   ei = (const int*)d_in[1];
    const float* W1 = (const float*)d_in[2];
    const float* b1 = (const float*)d_in[3];
    const float* W2 = (const float*)d_in[4];
    const float* b2 = (const float*)d_in[5];
    const float* W3 = (const float*)d_in[6];
    const float* b3 = (const float*)d_in[7];

    const int n = in_sizes[0] / 128;    // 50000 nodes
    const int E = in_sizes[1] / 2;      // 640000 edges
    const int* src = ei;
    const int* dst = ei + E;

    float* ws   = (float*)d_ws;
    float* deg  = ws;                               // n floats
    float* dinv = ws + n;                           // n floats
    float* bufA = ws + 2 * (size_t)n;               // n*128 floats (h)
    float* bufB = bufA + (size_t)n * 128;           // n*128 floats (agg / next x)
    float* out  = (float*)d_out;                    // n*64 floats

    const int B = 256;
    // degree + normalization (recomputed every call: deterministic, stateless)
    k_deg_init<<<(n + B - 1) / B, B, 0, stream>>>(deg, n);
    k_deg_count<<<(E + B - 1) / B, B, 0, stream>>>(dst, deg, E);
    k_dinv<<<(n + B - 1) / B, B, 0, stream>>>(deg, dinv, n);

    dim3 g128(n / 16, 128 / 64);   // NT=4 -> 16x64 strip per wave
    dim3 g64 (n / 16,  64 / 64);

    // ---- layer 1: h1 = x@W1 (bufA); agg = self+bias -> bufB; edges; relu
    k_gemm_wmma<128, 128, 4><<<g128, 32, 0, stream>>>(x, W1, bufA);
    k_self_init<128><<<(n * 32 + B - 1) / B, B, 0, stream>>>(bufA, dinv, b1, bufB, n);
    k_edge_agg<128><<<(E * 32 + B - 1) / B, B, 0, stream>>>(bufA, src, dst, dinv, bufB, E);
    k_relu<<<(n * 32 + B - 1) / B, B, 0, stream>>>(bufB, n * 32);

    // ---- layer 2: h2 = bufB@W2 (bufA); agg -> bufB; edges; relu
    k_gemm_wmma<128, 128, 4><<<g128, 32, 0, stream>>>(bufB, W2, bufA);
    k_self_init<128><<<(n * 32 + B - 1) / B, B, 0, stream>>>(bufA, dinv, b2, bufB, n);
    k_edge_agg<128><<<(E * 32 + B - 1) / B, B, 0, stream>>>(bufA, src, dst, dinv, bufB, E);
    k_relu<<<(n * 32 + B - 1) / B, B, 0, stream>>>(bufB, n * 32);

    // ---- layer 3: h3 = bufB@W3 (bufA, n*64); agg = self+bias -> d_out; edges
    k_gemm_wmma<128, 64, 4><<<g64, 32, 0, stream>>>(bufB, W3, bufA);
    k_self_init<64><<<(n * 16 + B - 1) / B, B, 0, stream>>>(bufA, dinv, b3, out, n);
    k_edge_agg<64><<<(E * 16 + B - 1) / B, B, 0, stream>>>(bufA, src, dst, dinv, out, E);
    // bias already folded into k_self_init; no epilogue for layer 3
}